// GATModel_39960375722435
// MI455X (gfx1250) — compile-verified
//
#include <hip/hip_runtime.h>
#include <math.h>
#include <stdint.h>

// ---------------------------------------------------------------------------
// GAT model for MI455X (gfx1250, wave32).
// Dims (from reference): N=65536 nodes, E=1M edges, G=512 graphs,
// F_in=128, H=8 heads, D=32, C=H*D=256, P=512.
// ---------------------------------------------------------------------------

typedef __attribute__((ext_vector_type(2))) float v2f;
typedef __attribute__((ext_vector_type(8))) float v8f;

#define NEG_SLOPE 0.2f
#define C_DIM 256      // H*D
#define H_DIM 8
#define D_DIM 32
#define P_DIM 512

// ---------------------------------------------------------------------------
// float atomic max via sign-split integer atomics
// (lowers to GLOBAL_ATOMIC_MAX_I32 / GLOBAL_ATOMIC_MIN_U32)
// ---------------------------------------------------------------------------
__device__ __forceinline__ void atomicMaxFloat(float* addr, float v) {
  if (v >= 0.0f) {
    atomicMax((int*)addr, __float_as_int(v));
  } else {
    atomicMin((unsigned int*)addr, __float_as_uint(v));
  }
}

__device__ __forceinline__ float leaky(float x) {
  return x > 0.0f ? x : NEG_SLOPE * x;
}

// ---------------------------------------------------------------------------
// WMMA fp32 GEMM: C[N, 256] = A[N, KIN] @ W[KIN, 256]
// Block: 256 threads (8 waves). Block owns 16 rows; wave w owns cols [32w,32w+32)
// as two 16x16 tiles. A tile staged in LDS (padded stride -> no bank conflicts).
//
// V_WMMA_F32_16X16X4_F32 operand layouts (ISA 7.12.2):
//   A 16x4 : lanes 0-15 row M=lane, VGPR0=K0,VGPR1=K1; lanes 16-31 K2/K3
//   B 4x16 : row striped across lanes within a VGPR (VGPR0: K0 | K2 halves,
//            VGPR1: K1 | K3 halves), N = lane&15
//   C 16x16: VGPR v -> row v (lanes 0-15) / row v+8 (lanes 16-31), N = lane&15
// ---------------------------------------------------------------------------
template <int KIN>
__global__ void gemm_wmma_kernel(const float* __restrict__ A,
                                 const float* __restrict__ W,
                                 float* __restrict__ C) {
  __shared__ float As[16 * (KIN + 1)];
  const int row0 = blockIdx.x * 16;
  const int tid = threadIdx.x;

  // cooperative, coalesced load of the 16 x KIN A tile
  for (int idx = tid; idx < 16 * KIN; idx += 256) {
    int r = idx / KIN;
    int c = idx % KIN;
    As[r * (KIN + 1) + c] = A[(row0 + r) * KIN + c];
  }
  __syncthreads();

  const int wave = tid >> 5;
  const int lane = tid & 31;
  const int laneHalf = lane >> 4;   // 0 or 1
  const int lane15 = lane & 15;
  const int col0 = wave * 32;

  v8f acc0 = {};
  v8f acc1 = {};

  for (int k0 = 0; k0 < KIN; k0 += 4) {
    v2f a;
    const float* ap = &As[lane15 * (KIN + 1) + k0 + laneHalf * 2];
    a.x = ap[0];
    a.y = ap[1];

    const float* wp = &W[(k0 + laneHalf * 2) * C_DIM];
    v2f b0, b1;
    b0.x = wp[col0 + lane15];
    b0.y = wp[C_DIM + col0 + lane15];
    b1.x = wp[col0 + 16 + lane15];
    b1.y = wp[C_DIM + col0 + 16 + lane15];

    acc0 = __builtin_amdgcn_wmma_f32_16x16x4_f32(false, a, false, b0,
                                                 (short)0, acc0, false, false);
    acc1 = __builtin_amdgcn_wmma_f32_16x16x4_f32(false, a, false, b1,
                                                 (short)0, acc1, false, false);
  }

  const int rbase = row0 + laneHalf * 8;
#pragma unroll
  for (int v = 0; v < 8; ++v) {
    C[(rbase + v) * C_DIM + col0 + lane15] = acc0[v];
    C[(rbase + v) * C_DIM + col0 + 16 + lane15] = acc1[v];
  }
}

// ---------------------------------------------------------------------------
// el[n,h] = sum_d h[n,h,d]*al[h,d] ; er likewise. One block (256 thr) per node;
// each wave32 = one head, lanes = 32 dims, wave shuffle reduction.
// ---------------------------------------------------------------------------
__global__ void eler_kernel(const float* __restrict__ h,
                            const float* __restrict__ al,
                            const float* __restrict__ ar,
                            float* __restrict__ el, float* __restrict__ er) {
  const int n = blockIdx.x;
  const int t = threadIdx.x;          // 0..255
  const int hh = t >> 5;
  const int d = t & 31;
  const float v = h[n * C_DIM + t];
  float pl = v * al[hh * D_DIM + d];
  float pr = v * ar[hh * D_DIM + d];
#pragma unroll
  for (int off = 16; off > 0; off >>= 1) {
    pl += __shfl_xor(pl, off, 32);
    pr += __shfl_xor(pr, off, 32);
  }
  if (d == 0) {
    el[n * H_DIM + hh] = pl;
    er[n * H_DIM + hh] = pr;
  }
}

__global__ void init_neg_inf_kernel(float* __restrict__ m, int count) {
  int i = blockIdx.x * blockDim.x + threadIdx.x;
  if (i < count) m[i] = -INFINITY;
}

// segment max of leaky_relu(el[src]+er[dst]) over dst, per head
__global__ void edge_max_kernel(const int* __restrict__ src,
                                const int* __restrict__ dst,
                                const float* __restrict__ el,
                                const float* __restrict__ er,
                                float* __restrict__ m, int E) {
  int t = blockIdx.x * blockDim.x + threadIdx.x;
  if (t >= E * H_DIM) return;
  int e = t >> 3;
  int hh = t & 7;
  int s = src[e], d = dst[e];
  float x = leaky(el[s * H_DIM + hh] + er[d * H_DIM + hh]);
  atomicMaxFloat(&m[d * H_DIM + hh], x);
}

__global__ void finalize_m_kernel(float* __restrict__ m, int count) {
  int i = blockIdx.x * blockDim.x + threadIdx.x;
  if (i < count) {
    float v = m[i];
    if (!isfinite(v)) m[i] = 0.0f;
  }
}

// ex[e,h] = exp(e_eh - m[dst]); denom[dst,h] += ex
__global__ void edge_exp_kernel(const int* __restrict__ src,
                                const int* __restrict__ dst,
                                const float* __restrict__ el,
                                const float* __restrict__ er,
                                const float* __restrict__ m,
                                float* __restrict__ exbuf,
                                float* __restrict__ denom, int E) {
  int t = blockIdx.x * blockDim.x + threadIdx.x;
  if (t >= E * H_DIM) return;
  int e = t >> 3;
  int hh = t & 7;
  int s = src[e], d = dst[e];
  float x = leaky(el[s * H_DIM + hh] + er[d * H_DIM + hh]);
  float ex = __expf(x - m[d * H_DIM + hh]);
  exbuf[t] = ex;
  atomicAdd(&denom[d * H_DIM + hh], ex);
}

// acc[dst, h, d] += ex[e,h] * h[src, h, d]  (unnormalized message aggregation)
// one block (256 threads = 256 channels) per edge; h table is L2-resident.
__global__ void edge_agg_kernel(const int* __restrict__ src,
                                const int* __restrict__ dst,
                                const float* __restrict__ exbuf,
                                const float* __restrict__ h,
                                float* __restrict__ acc) {
  const int e = blockIdx.x;
  const int c = threadIdx.x;
  const int s = src[e];
  const int d = dst[e];
  const float ex = exbuf[e * H_DIM + (c >> 5)];
  atomicAdd(&acc[d * C_DIM + c], ex * h[s * C_DIM + c]);
}

// out = acc / denom (0 where denom == 0, matching empty-segment semantics)
__global__ void normalize_kernel(float* __restrict__ acc,
                                 const float* __restrict__ denom, int N) {
  int t = blockIdx.x * blockDim.x + threadIdx.x;
  if (t >= N * C_DIM) return;
  int n = t >> 8;
  int hh = (t & 255) >> 5;
  float den = denom[n * H_DIM + hh];
  float v = acc[t];
  acc[t] = den > 0.0f ? v / den : 0.0f;
}

__global__ void pool_sum_kernel(const float* __restrict__ x,
                                const int* __restrict__ gid,
                                float* __restrict__ sums, int N) {
  int t = blockIdx.x * blockDim.x + threadIdx.x;
  if (t >= N * C_DIM) return;
  int n = t >> 8;
  int c = t & 255;
  atomicAdd(&sums[gid[n] * C_DIM + c], x[t]);
}

__global__ void pool_count_kernel(const int* __restrict__ gid,
                                  float* __restrict__ counts, int N) {
  int n = blockIdx.x * blockDim.x + threadIdx.x;
  if (n < N) atomicAdd(&counts[gid[n]], 1.0f);
}

// predictor MLP: one block (512 threads) per graph.
__global__ void mlp_kernel(const float* __restrict__ sums,
                           const float* __restrict__ counts,
                           const float* __restrict__ Wd1,
                           const float* __restrict__ bd1,
                           const float* __restrict__ Wd2,
                           const float* __restrict__ bd2,
                           float* __restrict__ out) {
  __shared__ float pooled[C_DIM];
  __shared__ float red[P_DIM];
  const int g = blockIdx.x;
  const int p = threadIdx.x;           // 0..511
  const float inv = 1.0f / fmaxf(counts[g], 1.0f);
  if (p < C_DIM) pooled[p] = sums[g * C_DIM + p] * inv;
  __syncthreads();
  float acc = bd1[p];
#pragma unroll 8
  for (int c = 0; c < C_DIM; ++c) acc = fmaf(pooled[c], Wd1[c * P_DIM + p], acc);
  red[p] = fmaxf(acc, 0.0f) * Wd2[p];
  __syncthreads();
  for (int s = P_DIM / 2; s > 0; s >>= 1) {
    if (p < s) red[p] += red[p + s];
    __syncthreads();
  }
  if (p == 0) out[g] = red[0] + bd2[0];
}

// ---------------------------------------------------------------------------
// host side
// ---------------------------------------------------------------------------
static inline size_t alignUp(size_t v, size_t a) { return (v + a - 1) & ~(a - 1); }

extern "C" void kernel_launch(void* const* d_in, const int* in_sizes, int n_in,
                              void* d_out, int out_size, void* d_ws, size_t ws_size,
                              hipStream_t stream) {
  const float* node_feats = (const float*)d_in[0];
  const int* src = (const int*)d_in[1];
  const int* dst = (const int*)d_in[2];
  const int* gid = (const int*)d_in[3];
  const float* W0 = (const float*)d_in[5];
  const float* al0 = (const float*)d_in[6];
  const float* ar0 = (const float*)d_in[7];
  const float* W1 = (const float*)d_in[8];
  const float* al1 = (const float*)d_in[9];
  const float* ar1 = (const float*)d_in[10];
  const float* Wd1 = (const float*)d_in[11];
  const float* bd1 = (const float*)d_in[12];
  const float* Wd2 = (const float*)d_in[13];
  const float* bd2 = (const float*)d_in[14];
  float* out = (float*)d_out;

  const int N = in_sizes[3];        // 65536
  const int E = in_sizes[1];        // 1048576
  const int G = out_size;           // 512 (output is [G,1])

  // workspace layout
  char* ws = (char*)d_ws;
  size_t off = 0;
  float* bufH = (float*)(ws + off);   off = alignUp(off + (size_t)N * C_DIM * 4, 256);
  float* bufA = (float*)(ws + off);   off = alignUp(off + (size_t)N * C_DIM * 4, 256);
  float* el   = (float*)(ws + off);   off = alignUp(off + (size_t)N * H_DIM * 4, 256);
  float* er   = (float*)(ws + off);   off = alignUp(off + (size_t)N * H_DIM * 4, 256);
  float* m    = (float*)(ws + off);   off = alignUp(off + (size_t)N * H_DIM * 4, 256);
  float* den  = (float*)(ws + off);   off = alignUp(off + (size_t)N * H_DIM * 4, 256);
  float* exb  = (float*)(ws + off);   off = alignUp(off + (size_t)E * H_DIM * 4, 256);
  float* sums = (float*)(ws + off);   off = alignUp(off + (size_t)G * C_DIM * 4, 256);
  float* cnts = (float*)(ws + off);   off = alignUp(off + (size_t)G * 4, 256);
  (void)ws_size; (void)n_in;

  const int NH = N * H_DIM;
  const int EH = E * H_DIM;
  const int nhBlocks = (NH + 255) / 256;
  const int ehBlocks = (EH + 255) / 256;
  const int ncBlocks = (N * C_DIM + 255) / 256;

  // ---------------- layer 0 ----------------
  gemm_wmma_kernel<128><<<N / 16, 256, 0, stream>>>(node_feats, W0, bufH);
  eler_kernel<<<N, 256, 0, stream>>>(bufH, al0, ar0, el, er);

  init_neg_inf_kernel<<<nhBlocks, 256, 0, stream>>>(m, NH);
  hipMemsetAsync(den, 0, (size_t)NH * 4, stream);
  hipMemsetAsync(bufA, 0, (size_t)N * C_DIM * 4, stream);

  edge_max_kernel<<<ehBlocks, 256, 0, stream>>>(src, dst, el, er, m, E);
  finalize_m_kernel<<<nhBlocks, 256, 0, stream>>>(m, NH);
  edge_exp_kernel<<<ehBlocks, 256, 0, stream>>>(src, dst, el, er, m, exb, den, E);
  edge_agg_kernel<<<E, 256, 0, stream>>>(src, dst, exb, bufH, bufA);
  normalize_kernel<<<ncBlocks, 256, 0, stream>>>(bufA, den, N);

  // ---------------- layer 1 ----------------
  gemm_wmma_kernel<256><<<N / 16, 256, 0, stream>>>(bufA, W1, bufH);
  eler_kernel<<<N, 256, 0, stream>>>(bufH, al1, ar1, el, er);

  init_neg_inf_kernel<<<nhBlocks, 256, 0, stream>>>(m, NH);
  hipMemsetAsync(den, 0, (size_t)NH * 4, stream);
  hipMemsetAsync(bufA, 0, (size_t)N * C_DIM * 4, stream);   // bufA now free (GEMM consumed it)

  edge_max_kernel<<<ehBlocks, 256, 0, stream>>>(src, dst, el, er, m, E);
  finalize_m_kernel<<<nhBlocks, 256, 0, stream>>>(m, NH);
  edge_exp_kernel<<<ehBlocks, 256, 0, stream>>>(src, dst, el, er, m, exb, den, E);
  edge_agg_kernel<<<E, 256, 0, stream>>>(src, dst, exb, bufH, bufA);
  normalize_kernel<<<ncBlocks, 256, 0, stream>>>(bufA, den, N);

  // ---------------- readout + MLP ----------------
  hipMemsetAsync(sums, 0, (size_t)G * C_DIM * 4, stream);
  hipMemsetAsync(cnts, 0, (size_t)G * 4, stream);
  pool_sum_kernel<<<ncBlocks, 256, 0, stream>>>(bufA, gid, sums, N);
  pool_count_kernel<<<(N + 255) / 256, 256, 0, stream>>>(gid, cnts, N);
  mlp_kernel<<<G, P_DIM, 0, stream>>>(sums, cnts, Wd1, bd1, Wd2, bd2, out);
}